// SNN_67937792688236
// MI455X (gfx1250) — compile-verified
//
#include <hip/hip_runtime.h>

// LIF scan: T=1024 steps over B*N = 32*2048 = 65536 independent neurons.
// Memory-bound streaming scan (~512 MiB @ 23.3 TB/s => ~22us floor).
// Strategy: deep async global->LDS pipeline (ASYNCcnt, 48 stages = ~12MB
// machine-wide in flight) hides HBM latency around the serial recurrence;
// time-unrolled x4 to amortize waits; NT b128 stores for spike output.

#define T_STEPS   1024
#define BN        65536           // B*N
#define BLOCK     64              // 2 waves per workgroup
#define COLS4     (BN / 4)        // 16384 float4 columns
#define GRID      (COLS4 / BLOCK) // 256 workgroups
#define DEPTH     48              // async pipeline depth (< ASYNCcnt max 63)
#define UNROLL_T  4               // timesteps per wait group (divides T)

#define DECAY_V   0.1f            // DT * TAU_MEM_INV
#define DECAY_I   0.8f            // 1 - DT * TAU_SYN_INV
#define V_TH_C    1.0f

typedef __attribute__((ext_vector_type(4))) float v4f;
// Matches the builtin's parameter type: 'int __attribute__((__vector_size__(16)))'
typedef int v4i_vs __attribute__((vector_size(16)));

#if defined(__HIP_DEVICE_COMPILE__) && __has_builtin(__builtin_amdgcn_global_load_async_to_lds_b128)
#define HAVE_ASYNC_LDS 1
// (global src AS1, lds dst AS3, imm offset, imm cpol)
#define ASYNC_CP_B128(gsrc, ldst)                                   \
    __builtin_amdgcn_global_load_async_to_lds_b128(                 \
        (__attribute__((address_space(1))) v4i_vs*)(gsrc),          \
        (__attribute__((address_space(3))) v4i_vs*)(ldst), 0, 0)
#else
#define HAVE_ASYNC_LDS 0
#endif

#if defined(__HIP_DEVICE_COMPILE__) && __has_builtin(__builtin_amdgcn_s_wait_asynccnt)
#define WAIT_ASYNC(n) __builtin_amdgcn_s_wait_asynccnt(n)
#else
#define WAIT_ASYNC(n) asm volatile("s_wait_asynccnt %0" ::"i"(n) : "memory")
#endif

#if defined(__HIP_DEVICE_COMPILE__) && __has_builtin(__builtin_amdgcn_s_wait_dscnt)
#define WAIT_DS(n) __builtin_amdgcn_s_wait_dscnt(n)
#else
#define WAIT_DS(n) asm volatile("s_wait_dscnt %0" ::"i"(n) : "memory")
#endif

#define COMPILER_MEM_BARRIER() asm volatile("" ::: "memory")

// One LIF update for one element. Returns spike z in {0,1}.
__device__ __forceinline__ float lif_step(float& v, float& i, float xin) {
    float v_dec = v + DECAY_V * ((0.0f - v) + i);  // leak + integrate current
    float i_dec = i * DECAY_I;                     // synaptic decay
    float z     = (v_dec > V_TH_C) ? 1.0f : 0.0f;  // heaviside(v_dec - v_th)
    v = (z != 0.0f) ? 0.0f : v_dec;                // reset to V_RESET = 0
    i = i_dec + xin;                               // inject input current
    return z;
}

// Vector elements can't bind to float&; shuttle through locals (SROA'd away).
__device__ __forceinline__ v4f lif_step4(v4f& v, v4f& i, v4f xv) {
    v4f zv;
    float a, b;
    a = v.x; b = i.x; zv.x = lif_step(a, b, xv.x); v.x = a; i.x = b;
    a = v.y; b = i.y; zv.y = lif_step(a, b, xv.y); v.y = a; i.y = b;
    a = v.z; b = i.z; zv.z = lif_step(a, b, xv.z); v.z = a; i.z = b;
    a = v.w; b = i.w; zv.w = lif_step(a, b, xv.w); v.w = a; i.w = b;
    return zv;
}

__global__ __launch_bounds__(BLOCK) void lif_scan_kernel(
    const float* __restrict__ x,    // (T, B*N)
    float* __restrict__ spikes,     // (T, B*N)
    float* __restrict__ v_out,      // (B*N)
    float* __restrict__ i_out,      // (B*N)
    int T)                          // assumed: T % UNROLL_T == 0, T >= DEPTH
{
    const int tid = threadIdx.x;
    const int g   = blockIdx.x * BLOCK + tid;   // float4 column id
    const size_t col = (size_t)g * 4;           // element offset, 16B aligned

    v4f v = {0.f, 0.f, 0.f, 0.f};
    v4f i = {0.f, 0.f, 0.f, 0.f};

#if HAVE_ASYNC_LDS
    // Per-wave private staging ring in LDS: DEPTH timesteps deep.
    __shared__ v4f ring[DEPTH][BLOCK];          // 48 * 64 * 16B = 48 KB

    // Prologue: fill the pipeline (DEPTH outstanding async loads).
    #pragma unroll
    for (int k = 0; k < DEPTH; ++k) {
        ASYNC_CP_B128(x + (size_t)k * BN + col, &ring[k][tid]);
    }

    int stage = 0;                              // ring stage of timestep t
    for (int t = 0; t < T; t += UNROLL_T) {
        // Async loads complete in issue order: <= DEPTH-UNROLL_T outstanding
        // means the UNROLL_T oldest stages (t .. t+U-1) have landed in LDS.
        WAIT_ASYNC(DEPTH - UNROLL_T);
        COMPILER_MEM_BARRIER();

        v4f xv[UNROLL_T];
        int st[UNROLL_T];
        #pragma unroll
        for (int u = 0; u < UNROLL_T; ++u) {
            st[u] = stage;
            xv[u] = ring[stage][tid];           // ds_load_b128 (pipelined)
            stage = (stage + 1 == DEPTH) ? 0 : stage + 1;
        }

        // Retire LDS reads into VGPRs before overwriting the slots.
        WAIT_DS(0);
        COMPILER_MEM_BARRIER();

        #pragma unroll
        for (int u = 0; u < UNROLL_T; ++u) {
            if (t + DEPTH + u < T) {
                ASYNC_CP_B128(x + (size_t)(t + DEPTH + u) * BN + col,
                              &ring[st[u]][tid]);
            }
        }

        #pragma unroll
        for (int u = 0; u < UNROLL_T; ++u) {
            v4f zv = lif_step4(v, i, xv[u]);
            // Spikes are write-once streaming data: NT, don't pollute L2.
            __builtin_nontemporal_store(
                zv, (v4f*)(spikes + (size_t)(t + u) * BN + col));
        }
    }
    WAIT_ASYNC(0);
#else
    // Fallback: register software pipeline (8 NT loads in flight per wave).
    const int U = 8;
    for (int t = 0; t < T; t += U) {
        v4f xs[U];
        #pragma unroll
        for (int u = 0; u < U; ++u)
            xs[u] = __builtin_nontemporal_load(
                (const v4f*)(x + (size_t)(t + u) * BN + col));
        #pragma unroll
        for (int u = 0; u < U; ++u) {
            v4f zv = lif_step4(v, i, xs[u]);
            __builtin_nontemporal_store(
                zv, (v4f*)(spikes + (size_t)(t + u) * BN + col));
        }
    }
#endif

    // Final state (B*N each) — tiny, regular stores.
    *(v4f*)(v_out + col) = v;
    *(v4f*)(i_out + col) = i;
}

extern "C" void kernel_launch(void* const* d_in, const int* in_sizes, int n_in,
                              void* d_out, int out_size, void* d_ws, size_t ws_size,
                              hipStream_t stream) {
    (void)n_in; (void)d_ws; (void)ws_size; (void)out_size;
    const float* x = (const float*)d_in[0];
    const int T = in_sizes[0] / BN;             // 1024

    float* out    = (float*)d_out;
    float* spikes = out;                        // (T, BN)
    float* v_out  = out + (size_t)T * BN;       // (BN)
    float* i_out  = v_out + BN;                 // (BN)

    lif_scan_kernel<<<GRID, BLOCK, 0, stream>>>(x, spikes, v_out, i_out, T);
}